// Embedding_62843961475697
// MI455X (gfx1250) — compile-verified
//
#include <hip/hip_runtime.h>
#include <hip/hip_bf16.h>
#include <math.h>

// Embedding gather + bias + sinusoidal positional encoding.
//   out[b,s,d] = W[d, x[b,s]] + bias[d] + pe[s,d]
// B=8, S=2048, D=512, VOCAB=50257. Output: float32 [8,2048,512].
//
// Memory-bound: ~32MB gathered reads (W = 103MB fits in 192MB L2) + 32MB
// streaming writes -> ~3-6us at 23.3 TB/s. WMMA is inapplicable (one-hot GEMM
// would be ~8e11 FLOPs of wasted work). CDNA5 paths used:
//   - global_load_async_to_lds_b128 + s_wait_asynccnt for bias staging
//   - non-temporal b128 output stores (keep W resident in L2, stream output)
//   - wave32-sized blocks, coalesced 128-bit stores

#define D_MODEL 512
#define VOCAB_N 50257
#define BATCH_N 8
#define SEQ_N   2048
#define THREADS 128   // 4 wave32s; thread t owns d = 4t..4t+3

typedef __attribute__((ext_vector_type(4))) int   v4i;
typedef __attribute__((ext_vector_type(4))) float v4f;
typedef v4i __attribute__((address_space(1)))* gptr_v4i;  // global
typedef v4i __attribute__((address_space(3)))* lptr_v4i;  // LDS

__global__ __launch_bounds__(THREADS)
void embed_pe_kernel(const int* __restrict__ x,
                     const float* __restrict__ W,
                     const float* __restrict__ bias,
                     float* __restrict__ out)
{
    const int s  = blockIdx.x;        // sequence position
    const int t  = threadIdx.x;       // 0..127
    const int d0 = t << 2;            // first of 4 consecutive dims

    __shared__ int   s_tok[BATCH_N];
    __shared__ float s_bias[D_MODEL];

    // ---- Stage bias row into LDS via CDNA5 async copy (ASYNCcnt path) ----
#if __has_builtin(__builtin_amdgcn_global_load_async_to_lds_b128)
    __builtin_amdgcn_global_load_async_to_lds_b128(
        (gptr_v4i)(bias + d0),
        (lptr_v4i)(&s_bias[d0]),
        /*offset=*/0, /*cpol=*/0);
#else
    *(v4f*)(&s_bias[d0]) = *(const v4f*)(bias + d0);
#endif

    // Tokens for this sequence position across the 8 batch rows.
    if (t < BATCH_N) s_tok[t] = x[t * SEQ_N + s];

#if __has_builtin(__builtin_amdgcn_s_wait_asynccnt)
    __builtin_amdgcn_s_wait_asynccnt(0);
#else
    asm volatile("s_wait_asynccnt 0" ::: "memory");
#endif
    __syncthreads();

    // ---- Positional encoding for this thread's 4 dims (registers) ----
    // even d: cos((s+1) / 10000^(d/512));  odd d: sin((s+1) / 10000^((d+1)/512))
    // For d0 even: exponents used are d0/512, (d0+2)/512 (shared by d0+1,d0+2),
    // and (d0+4)/512.
    const float p       = (float)(s + 1);
    const float L2_10K  = 13.287712379549449f;   // log2(10000)
    const float inv0 = exp2f(-(float)(d0    ) * (1.0f / 512.0f) * L2_10K);
    const float inv1 = exp2f(-(float)(d0 + 2) * (1.0f / 512.0f) * L2_10K);
    const float inv2 = exp2f(-(float)(d0 + 4) * (1.0f / 512.0f) * L2_10K);

    const float pepb0 = cosf(p * inv0) + s_bias[d0 + 0];
    const float pepb1 = sinf(p * inv1) + s_bias[d0 + 1];
    const float pepb2 = cosf(p * inv1) + s_bias[d0 + 2];
    const float pepb3 = sinf(p * inv2) + s_bias[d0 + 3];

    // Column pointers into W (row-major [D, VOCAB]); gather index = token id.
    const float* __restrict__ Wc0 = W + (size_t)(d0 + 0) * VOCAB_N;
    const float* __restrict__ Wc1 = W + (size_t)(d0 + 1) * VOCAB_N;
    const float* __restrict__ Wc2 = W + (size_t)(d0 + 2) * VOCAB_N;
    const float* __restrict__ Wc3 = W + (size_t)(d0 + 3) * VOCAB_N;

    // ---- 8 batch rows: 4 strided gather loads + one coalesced NT b128 store ----
#pragma unroll
    for (int bb = 0; bb < BATCH_N; ++bb) {
        const int tok = s_tok[bb];
        v4f v;
        v.x = Wc0[tok] + pepb0;
        v.y = Wc1[tok] + pepb1;
        v.z = Wc2[tok] + pepb2;
        v.w = Wc3[tok] + pepb3;
        v4f* dst = (v4f*)(out + ((size_t)bb * SEQ_N + s) * D_MODEL + d0);
        // Output is write-once: stream past L2 (NT) so W stays L2-resident.
        __builtin_nontemporal_store(v, dst);
    }
}

extern "C" void kernel_launch(void* const* d_in, const int* in_sizes, int n_in,
                              void* d_out, int out_size, void* d_ws, size_t ws_size,
                              hipStream_t stream) {
    const int*   x    = (const int*)  d_in[0];   // [8, 2048] token ids
    const float* W    = (const float*)d_in[1];   // [512, 50257]
    const float* bias = (const float*)d_in[2];   // [512]
    float*       out  = (float*)d_out;           // [8, 2048, 512]

    embed_pe_kernel<<<SEQ_N, THREADS, 0, stream>>>(x, W, bias, out);
}